// PatchGCN_27101243638336
// MI455X (gfx1250) — compile-verified
//
#include <hip/hip_runtime.h>
#include <hip/hip_bf16.h>

typedef __attribute__((ext_vector_type(2))) float v2f;
typedef __attribute__((ext_vector_type(8))) float v8f;

// ---------------------------------------------------------------- utilities
__global__ void fill_zero_kernel(float* __restrict__ p, long n) {
    long i = (long)blockIdx.x * blockDim.x + threadIdx.x;
    long stride = (long)gridDim.x * blockDim.x;
    for (; i < n; i += stride) p[i] = 0.0f;
}

__global__ void count_deg_kernel(const int* __restrict__ src, const int* __restrict__ dst,
                                 float* __restrict__ deg_out, float* __restrict__ deg_in, int E) {
    int i = blockIdx.x * blockDim.x + threadIdx.x;
    if (i < E) {
        atomicAdd(&deg_out[src[i]], 1.0f);   // integer-valued: order-independent
        atomicAdd(&deg_in[dst[i]], 1.0f);
    }
}

__global__ void finalize_deg_kernel(float* __restrict__ p, long n) {
    long i = (long)blockIdx.x * blockDim.x + threadIdx.x;
    if (i < n) p[i] = rsqrtf(fmaxf(p[i], 1.0f));
}

// out[n,d] = in[n,d] * norm[n]   (D fixed = 128)
__global__ void scale_rows128_kernel(const float* __restrict__ in, const float* __restrict__ norm,
                                     float* __restrict__ out, long total) {
    long i = (long)blockIdx.x * blockDim.x + threadIdx.x;
    if (i < total) out[i] = in[i] * norm[i >> 7];
}

// h[n,d] = leaky( agg[n,d]*norm_in[n] + b[d] ), D = 128
__global__ void epilogue128_kernel(const float* __restrict__ agg, const float* __restrict__ norm,
                                   const float* __restrict__ b, float* __restrict__ out, long total) {
    long i = (long)blockIdx.x * blockDim.x + threadIdx.x;
    if (i < total) {
        float v = agg[i] * norm[i >> 7] + b[i & 127];
        out[i] = v > 0.0f ? v : 0.01f * v;
    }
}

// ------------------------------------------------------- SpMM (scatter-add)
// one block per edge, 128 threads = 128 feature dims; coalesced gather +
// global_atomic_add_f32 scatter (L2-resident: node tensors are 25 MB << 192 MB L2)
__global__ void spmm128_kernel(const float* __restrict__ x, const int* __restrict__ src,
                               const int* __restrict__ dst, float* __restrict__ agg, int E) {
    int e = blockIdx.x;
    if (e >= E) return;
    int s = src[e], d = dst[e];
    int t = threadIdx.x;
    atomicAdd(&agg[(long)d * 128 + t], x[(long)s * 128 + t]);
}

// ------------------------------------------------------- WMMA fp32 GEMM
// C[N,Dout] = leaky( (rowScale .* A[N,K]) @ W[K,Dout] + bias )
// Block = 128 threads (4 wave32), one 16-row node tile staged (pre-scaled) in LDS.
// Requires N % 16 == 0, K % 4 == 0, Dout % 16 == 0 (true here: 50000=16*3125).
__global__ void gemm_wmma_f32_kernel(const float* __restrict__ A,
                                     const float* __restrict__ rowScale,   // nullable
                                     const float* __restrict__ W,
                                     const float* __restrict__ bias,       // nullable
                                     float* __restrict__ C,
                                     int K, int Dout, int leaky) {
    __shared__ float tile[16 * 256];              // 16 rows x K (K <= 256) = 16 KB
    const int rowBase = blockIdx.x * 16;

    // cooperative stage: tile[r*K+k] = A[rowBase+r, k] * rowScale[rowBase+r]
    for (int i = threadIdx.x; i < 16 * K; i += 128) {
        int r = i / K;
        int k = i - r * K;
        int row = rowBase + r;
        float sc = rowScale ? rowScale[row] : 1.0f;
        tile[i] = A[(long)row * K + k] * sc;
    }
    __syncthreads();

    const int lane = threadIdx.x & 31;
    const int half = lane >> 4;                   // K-pair selector (A) / row-half (C)
    const int m    = lane & 15;                   // A row / B,C column within tile
    const int wave = threadIdx.x >> 5;            // 0..3

    for (int ct = wave; ct < (Dout >> 4); ct += 4) {
        const int col = (ct << 4) + m;
        v8f c = {0.f, 0.f, 0.f, 0.f, 0.f, 0.f, 0.f, 0.f};
        for (int k0 = 0; k0 < K; k0 += 4) {
            const int ka = k0 + (half << 1);
            v2f a, b;
            // A 16x4 layout: lanes 0-15 hold K=k0,k0+1; lanes 16-31 hold K=k0+2,k0+3
            a.x = tile[m * K + ka];
            a.y = tile[m * K + ka + 1];
            // B 4x16 layout (row striped across lanes, lane-half selects K-pair)
            b.x = W[(long)ka * Dout + col];
            b.y = W[(long)(ka + 1) * Dout + col];
            c = __builtin_amdgcn_wmma_f32_16x16x4_f32(
                    false, a, false, b, (short)0, c, false, false);
        }
        const float bv = bias ? bias[col] : 0.0f;
#pragma unroll
        for (int v = 0; v < 8; ++v) {
            float val = c[v] + bv;
            if (leaky) val = val > 0.0f ? val : 0.01f * val;
            // C/D layout: VGPR v -> row v (lanes 0-15) / v+8 (lanes 16-31)
            C[(long)(rowBase + v + (half << 3)) * Dout + col] = val;
        }
    }
}

// ------------------------------------------------------- mean over nodes
// one block per output dim; deterministic tree reduction (no float atomics on out)
__global__ void mean_nodes128_kernel(const float* __restrict__ h, float* __restrict__ out, int N) {
    __shared__ float red[256];
    const int d = blockIdx.x;                     // 0..127
    float s = 0.0f;
    for (int n = threadIdx.x; n < N; n += 256) s += h[(long)n * 128 + d];
    red[threadIdx.x] = s;
    __syncthreads();
    for (int off = 128; off > 0; off >>= 1) {
        if (threadIdx.x < off) red[threadIdx.x] += red[threadIdx.x + off];
        __syncthreads();
    }
    if (threadIdx.x == 0) out[d] = red[0] / (float)N;
}

// ---------------------------------------------------------------- driver
extern "C" void kernel_launch(void* const* d_in, const int* in_sizes, int n_in,
                              void* d_out, int out_size, void* d_ws, size_t ws_size,
                              hipStream_t stream) {
    const float* n_feat = (const float*)d_in[0];
    const int*   src    = (const int*)d_in[1];
    const int*   dst    = (const int*)d_in[2];
    const float* W0 = (const float*)d_in[3];
    const float* b0 = (const float*)d_in[4];
    const float* W1 = (const float*)d_in[5];
    const float* b1 = (const float*)d_in[6];
    const float* W2 = (const float*)d_in[7];
    const float* b2 = (const float*)d_in[8];
    float* out = (float*)d_out;

    const int N = in_sizes[0] / 128;              // 50000
    const int E = in_sizes[1];                    // 800000
    const int nodeTiles = N / 16;                 // exact: 50000 = 16*3125

    // workspace layout (floats)
    float* ws = (float*)d_ws;
    float* norm_out = ws;                         // [N]
    float* norm_in  = ws + N;                     // [N]
    size_t off = ((size_t)2 * N + 63) & ~(size_t)63;
    float* bufA = ws + off;                       // [N,256]
    float* bufB = bufA + (size_t)N * 256;         // [N,256]

    const int T = 256;
    auto blk = [](long n, int t) { return (int)((n + t - 1) / t); };
    const long n128 = (long)N * 128;

    // degrees -> rsqrt norms
    fill_zero_kernel<<<blk(2L * N, T), T, 0, stream>>>(norm_out, 2L * N);
    count_deg_kernel<<<blk(E, T), T, 0, stream>>>(src, dst, norm_out, norm_in, E);
    finalize_deg_kernel<<<blk(2L * N, T), T, 0, stream>>>(norm_out, 2L * N);

    // ---- layer 0 (128 -> 256): scale, aggregate, then GEMM (din < dout)
    scale_rows128_kernel<<<blk(n128, T), T, 0, stream>>>(n_feat, norm_out, bufA, n128);
    fill_zero_kernel<<<blk(n128, T), T, 0, stream>>>(bufB, n128);
    spmm128_kernel<<<E, 128, 0, stream>>>(bufA, src, dst, bufB, E);
    gemm_wmma_f32_kernel<<<nodeTiles, 128, 0, stream>>>(bufB, norm_in, W0, b0, bufA,
                                                        128, 256, /*leaky=*/1);   // h1 = bufA [N,256]

    // ---- layer 1 (256 -> 128): GEMM first (din > dout), then aggregate
    gemm_wmma_f32_kernel<<<nodeTiles, 128, 0, stream>>>(bufA, norm_out, W1, nullptr, bufB,
                                                        256, 128, /*leaky=*/0);   // xs2 = bufB [N,128]
    fill_zero_kernel<<<blk(n128, T), T, 0, stream>>>(bufA, n128);
    spmm128_kernel<<<E, 128, 0, stream>>>(bufB, src, dst, bufA, E);
    epilogue128_kernel<<<blk(n128, T), T, 0, stream>>>(bufA, norm_in, b1, bufB, n128); // h2 = bufB

    // ---- layer 2 (128 -> 128): scale, aggregate, then GEMM
    scale_rows128_kernel<<<blk(n128, T), T, 0, stream>>>(bufB, norm_out, bufA, n128);
    fill_zero_kernel<<<blk(n128, T), T, 0, stream>>>(bufB, n128);
    spmm128_kernel<<<E, 128, 0, stream>>>(bufA, src, dst, bufB, E);
    gemm_wmma_f32_kernel<<<nodeTiles, 128, 0, stream>>>(bufB, norm_in, W2, b2, bufA,
                                                        128, 128, /*leaky=*/1);   // h3 = bufA

    // ---- mean over nodes -> [1,128]
    mean_nodes128_kernel<<<128, 256, 0, stream>>>(bufA, out, N);
}